// YoloNmsLayer_58823872086351
// MI455X (gfx1250) — compile-verified
//
#include <hip/hip_runtime.h>
#include <stdint.h>

// Problem constants (match reference)
#define BB 8
#define NN 10647
#define CC 80
#define MM 100
#define IOU_TH 0.5f
#define SCORE_TH 0.3f

#define TPB 1024          // 32 wave32s per block
#define CH  11            // ceil(NN / TPB)
#define CH2 8             // ceil(CC*MM / TPB) = ceil(8000/1024)

// ---- CDNA5 async global->LDS path (guarded; device pass only) -------------
#if defined(__HIP_DEVICE_COMPILE__)
#if __has_builtin(__builtin_amdgcn_global_load_async_to_lds_b32)
#define HAVE_ASYNC_LDS 1
#endif
#endif

#ifdef HAVE_ASYNC_LDS
typedef __attribute__((address_space(1))) int g_int;
typedef __attribute__((address_space(3))) int l_int;

__device__ __forceinline__ void async_g2l_b32(const float* g, float* l) {
  // global (as1) 64-bit address == flat address; LDS offset == low 32 bits of
  // the flat shared address (CDNA5 aperture places LDS offset in addr[31:0]).
  __builtin_amdgcn_global_load_async_to_lds_b32(
      (g_int*)(uintptr_t)(const void*)g,
      (l_int*)(uint32_t)(uintptr_t)(void*)l,
      0, 0);
}
__device__ __forceinline__ void wait_async0() {
#if __has_builtin(__builtin_amdgcn_s_wait_asynccnt)
  __builtin_amdgcn_s_wait_asynccnt(0);
#else
  asm volatile("s_wait_asynccnt 0" ::: "memory");
#endif
}
#endif

// ---- wave32 / block argmax with first-index tie-break ---------------------
__device__ __forceinline__ void wave_argmax(float& s, int& i) {
#pragma unroll
  for (int off = 16; off > 0; off >>= 1) {
    float os = __shfl_down(s, off, 32);
    int   oi = __shfl_down(i, off, 32);
    if (os > s || (os == s && oi < i)) { s = os; i = oi; }
  }
}

// Requires blockDim.x == 1024 (exactly 32 waves).
__device__ __forceinline__ void block_argmax(float ls, int li,
                                             float* red_s, int* red_i,
                                             float* bc_s, int* bc_i) {
  wave_argmax(ls, li);
  const int lane = threadIdx.x & 31;
  const int wid  = threadIdx.x >> 5;
  if (lane == 0) { red_s[wid] = ls; red_i[wid] = li; }
  __syncthreads();                       // barrier A
  if (wid == 0) {
    float s2 = red_s[lane];
    int   i2 = red_i[lane];
    wave_argmax(s2, i2);
    if (lane == 0) { *bc_s = s2; *bc_i = i2; }
  }
  __syncthreads();                       // barrier B
}

// ---- Kernel 1: greedy NMS for one (batch, class) per block ----------------
__global__ __launch_bounds__(TPB)
void yolo_nms_perclass(const float* __restrict__ boxes,   // [B,N,4]
                       const float* __restrict__ conf,    // [B,N,1]
                       const float* __restrict__ probs,   // [B,N,C]
                       float* __restrict__ cscore,        // [B*C, M]
                       float* __restrict__ cbox) {        // [B*C, M, 4]
  __shared__ float stage[CH * TPB];      // prob staging (async path); 45 KB
  __shared__ float red_s[32];
  __shared__ int   red_i[32];
  __shared__ float bc_s;
  __shared__ int   bc_i;
  __shared__ float bc_box[4];

  const int bc = blockIdx.x;
  const int b  = bc / CC;
  const int c  = bc % CC;
  const int t  = threadIdx.x;

  const float* boxB  = boxes + (size_t)b * NN * 4;
  const float* confB = conf  + (size_t)b * NN;
  const float* probB = probs + (size_t)b * NN * CC + c;

  // Hint the box stream into cache (global_prefetch_b8).
  __builtin_prefetch(boxB + 4 * (size_t)t, 0, 0);

#ifdef HAVE_ASYNC_LDS
  // Strided class-prob gather via async global->LDS (ASYNCcnt path).
#pragma unroll
  for (int k = 0; k < CH; ++k) {
    int n = t + k * TPB;
    if (n < NN) async_g2l_b32(probB + (size_t)n * CC, &stage[n]);
  }
  wait_async0();   // each thread reads back only addresses it wrote
#endif

  // Per-thread registers: 11 anchors each (score, box, area)
  float sc[CH], x1[CH], y1[CH], x2[CH], y2[CH], ar[CH];
#pragma unroll
  for (int k = 0; k < CH; ++k) {
    int n = t + k * TPB;
    if (n < NN) {
      float4 bb = *(const float4*)(boxB + 4 * (size_t)n);
#ifdef HAVE_ASYNC_LDS
      float p = stage[n];
#else
      float p = probB[(size_t)n * CC];
#endif
      float s = confB[n] * p;
      sc[k] = (s > SCORE_TH) ? s : -1.0f;
      x1[k] = bb.x; y1[k] = bb.y; x2[k] = bb.z; y2[k] = bb.w;
      ar[k] = (bb.z - bb.x) * (bb.w - bb.y);
    } else {
      sc[k] = -1.0f;
      x1[k] = y1[k] = x2[k] = y2[k] = ar[k] = 0.0f;
    }
  }
  __syncthreads();  // stage buffer quiescent before reduction reuse of LDS

  int m;
  for (m = 0; m < MM; ++m) {
    // local register argmax (ascending n => strict '>' keeps smallest index)
    float ls = -2.0f; int li = 0x7fffffff;
#pragma unroll
    for (int k = 0; k < CH; ++k) {
      int n = t + k * TPB;
      if (sc[k] > ls) { ls = sc[k]; li = n; }
    }
    block_argmax(ls, li, red_s, red_i, &bc_s, &bc_i);
    const float swin = bc_s;
    const int   iwin = bc_i;
    if (swin <= 0.0f) break;             // uniform: bc_s identical block-wide

    if ((iwin & (TPB - 1)) == t) {       // owner thread publishes winner
      int k = iwin >> 10;
      bc_box[0] = x1[k]; bc_box[1] = y1[k];
      bc_box[2] = x2[k]; bc_box[3] = y2[k];
      size_t o = (size_t)bc * MM + m;
      cscore[o]       = swin;
      cbox[4 * o + 0] = x1[k];
      cbox[4 * o + 1] = y1[k];
      cbox[4 * o + 2] = x2[k];
      cbox[4 * o + 3] = y2[k];
      sc[k] = -1.0f;
    }
    __syncthreads();                     // barrier C: bc_box visible

    const float wx1 = bc_box[0], wy1 = bc_box[1];
    const float wx2 = bc_box[2], wy2 = bc_box[3];
    const float war = (wx2 - wx1) * (wy2 - wy1);
#pragma unroll
    for (int k = 0; k < CH; ++k) {
      float ix = fminf(wx2, x2[k]) - fmaxf(wx1, x1[k]);
      float iy = fminf(wy2, y2[k]) - fmaxf(wy1, y1[k]);
      float inter = fmaxf(ix, 0.0f) * fmaxf(iy, 0.0f);
      // iou > 0.5  <=>  inter/(war+ar-inter+1e-9) > 0.5  <=>  3*inter > war+ar+1e-9
      if (3.0f * inter > war + ar[k] + 1e-9f) sc[k] = -1.0f;
    }
  }

  // zero-fill remaining slots (matches reference's where(valid, ..., 0))
  for (int j = m + t; j < MM; j += TPB) {
    size_t o = (size_t)bc * MM + j;
    cscore[o] = 0.0f;
    cbox[4 * o + 0] = 0.0f; cbox[4 * o + 1] = 0.0f;
    cbox[4 * o + 2] = 0.0f; cbox[4 * o + 3] = 0.0f;
  }
}

// ---- Kernel 2: per-batch top-100 merge across 80*100 candidates -----------
__global__ __launch_bounds__(TPB)
void yolo_merge(const float* __restrict__ cscore,   // [B, C*M]
                const float* __restrict__ cbox,     // [B, C*M, 4]
                float* __restrict__ out) {
  __shared__ float red_s[32];
  __shared__ int   red_i[32];
  __shared__ float bc_s;
  __shared__ int   bc_i;

  const int b = blockIdx.x;
  const int t = threadIdx.x;
  const int NC = CC * MM;                // 8000

  float s[CH2];
#pragma unroll
  for (int k = 0; k < CH2; ++k) {
    int n = t + k * TPB;
    s[k] = (n < NC) ? cscore[(size_t)b * NC + n] : -2.0f;
  }

  float* sel_b = out + (size_t)b * MM * 4;
  float* sel_s = out + (size_t)BB * MM * 4 + (size_t)b * MM;
  float* sel_c = out + (size_t)BB * MM * 5 + (size_t)b * MM;
  int*   nval  = (int*)out + (size_t)BB * MM * 6;

  int cnt = 0;
  for (int m = 0; m < MM; ++m) {
    float ls = -3.0f; int li = 0x7fffffff;
#pragma unroll
    for (int k = 0; k < CH2; ++k) {
      int n = t + k * TPB;
      if (s[k] > ls) { ls = s[k]; li = n; }
    }
    block_argmax(ls, li, red_s, red_i, &bc_s, &bc_i);
    const float sw = bc_s;
    const int   iw = bc_i;
    if (sw > 0.0f) cnt++;                // identical in every thread
    if ((iw & (TPB - 1)) == t) {         // owner emits selection m
      int k = iw >> 10;
      s[k] = -3.0f;
      size_t src = (size_t)b * NC + iw;
      float4 bb = *(const float4*)(cbox + 4 * src);
      ((float4*)sel_b)[m] = bb;          // zeros for invalid (stored zeroed)
      sel_s[m] = (sw > 0.0f) ? sw : 0.0f;
      sel_c[m] = (sw > 0.0f) ? (float)(iw / MM) : 0.0f;
    }
  }
  if (t == 0) nval[b] = cnt;
}

// ---- host-side launcher ----------------------------------------------------
extern "C" void kernel_launch(void* const* d_in, const int* in_sizes, int n_in,
                              void* d_out, int out_size, void* d_ws, size_t ws_size,
                              hipStream_t stream) {
  (void)in_sizes; (void)n_in; (void)out_size; (void)ws_size;
  const float* boxes = (const float*)d_in[0];   // [B,N,4]
  const float* conf  = (const float*)d_in[1];   // [B,N,1]
  const float* probs = (const float*)d_in[2];   // [B,N,C]

  float* cscore = (float*)d_ws;                         // B*C*M floats
  float* cbox   = cscore + (size_t)BB * CC * MM;        // B*C*M*4 floats

  yolo_nms_perclass<<<dim3(BB * CC), dim3(TPB), 0, stream>>>(
      boxes, conf, probs, cscore, cbox);
  yolo_merge<<<dim3(BB), dim3(TPB), 0, stream>>>(cscore, cbox, (float*)d_out);
}